// spike_softmax_77360950935785
// MI455X (gfx1250) — compile-verified
//
#include <hip/hip_runtime.h>
#include <cstdint>
#include <cstddef>

// Masked softmax over rows of 2048 f32 (spike-train softmax collapses to this).
// One row per 256-thread block (8 waves, wave32). Row is staged into LDS with
// CDNA5 async global->LDS copies (ASYNCcnt) using non-temporal hints (data is
// streamed once, 512 MB >> 192 MB L2), reductions are wave32 shuffles + small
// LDS combine, and results go out with non-temporal b128 stores.

constexpr int ROW = 2048;           // softmax axis length
constexpr int TPB = 256;            // 8 waves of 32
constexpr int PER = ROW / TPB;      // 8 floats per thread
constexpr int NWAVE = TPB / 32;     // 8

typedef float f32x4 __attribute__((ext_vector_type(4)));

__global__ __launch_bounds__(TPB)
void spike_softmax_77360950935785_kernel(const float* __restrict__ X,
                                         float* __restrict__ Y) {
    __shared__ __align__(16) float tile[ROW];   // 8 KB row staging
    __shared__ float redbuf[NWAVE];             // cross-wave partials

    const int tid = threadIdx.x;
    const size_t rowbase = (size_t)blockIdx.x * (size_t)ROW;
    const float* __restrict__ src = X + rowbase;
    float*       __restrict__ dst = Y + rowbase;

    // ---- Stage row into LDS: CDNA5 async copy path (GV mode, ASYNCcnt) ----
#if defined(__gfx1250__)
    {
        // Low 32 bits of a generic shared pointer == wave-relative LDS byte
        // address (aperture layout), exactly what VDST of the async op wants.
        uint32_t lds0 = (uint32_t)(uintptr_t)(&tile[0]) + (uint32_t)(tid * 16);
        uint64_t g0   = (uint64_t)(uintptr_t)src + (uint64_t)(tid * 16);
        asm volatile("global_load_async_to_lds_b128 %0, %1, off th:TH_LOAD_NT"
                     :: "v"(lds0), "v"(g0) : "memory");
        uint32_t lds1 = lds0 + 4096u;
        uint64_t g1   = g0 + 4096u;
        asm volatile("global_load_async_to_lds_b128 %0, %1, off th:TH_LOAD_NT"
                     :: "v"(lds1), "v"(g1) : "memory");
#if __has_builtin(__builtin_amdgcn_s_wait_asynccnt)
        __builtin_amdgcn_s_wait_asynccnt(0);
#else
        asm volatile("s_wait_asynccnt 0x0" ::: "memory");
#endif
    }
#else
    {   // non-gfx1250 fallback: plain vectorized staging
        float4 a = ((const float4*)src)[tid];
        float4 b = ((const float4*)src)[tid + TPB];
        ((float4*)tile)[tid]       = a;
        ((float4*)tile)[tid + TPB] = b;
    }
#endif
    __syncthreads();   // every wave has drained its own ASYNCcnt -> tile valid

    // ---- Pull this thread's 8 elements into registers (ds_load_b128 x2) ----
    float x[PER];
    {
        float4 a = ((const float4*)tile)[tid];        // floats [4*tid .. 4*tid+3]
        float4 b = ((const float4*)tile)[tid + TPB];  // floats [1024+4*tid ..]
        x[0]=a.x; x[1]=a.y; x[2]=a.z; x[3]=a.w;
        x[4]=b.x; x[5]=b.y; x[6]=b.z; x[7]=b.w;
    }

    // ---- Row max: local -> wave32 shuffle -> cross-wave LDS combine ----
    float m = fmaxf(fmaxf(fmaxf(x[0], x[1]), fmaxf(x[2], x[3])),
                    fmaxf(fmaxf(x[4], x[5]), fmaxf(x[6], x[7])));
#pragma unroll
    for (int off = 16; off > 0; off >>= 1)
        m = fmaxf(m, __shfl_xor(m, off, 32));
    if ((tid & 31) == 0) redbuf[tid >> 5] = m;
    __syncthreads();
    float rowmax = redbuf[0];
#pragma unroll
    for (int i = 1; i < NWAVE; ++i) rowmax = fmaxf(rowmax, redbuf[i]);
    __syncthreads();   // redbuf about to be reused

    // ---- exp with mask + clamp; partition replicates the K=4 spike sum ----
    float e4[PER];
    float psum = 0.0f;
#pragma unroll
    for (int i = 0; i < PER; ++i) {
        float tp = fminf(x[i] - rowmax + 0.1f, 10.0f);  // tp[tp>10]=10
        float e  = (tp > -20.0f) ? expf(tp) : 0.0f;     // mask: no spikes below -20
        float q  = e * 0.25f;                            // one spike slice (K=4)
        e4[i] = q;
        float pc = (q + q) + q;                          // ((e/4+e/4)+e/4)+e/4
        pc = pc + q;
        psum += pc;
    }
#pragma unroll
    for (int off = 16; off > 0; off >>= 1)
        psum += __shfl_xor(psum, off, 32);
    if ((tid & 31) == 0) redbuf[tid >> 5] = psum;
    __syncthreads();
    float part = redbuf[0];
#pragma unroll
    for (int i = 1; i < NWAVE; ++i) part += redbuf[i];

    const float pinv = 1.0f / part;   // exact reciprocal (invert_tensor_precise)

    // ---- MAC of spike train against p_inv over K steps, then NT store ----
    float o[PER];
#pragma unroll
    for (int i = 0; i < PER; ++i) {
        float t = e4[i] * pinv;
        float s = (t + t) + t;        // ((t+t)+t)+t
        s = s + t;
        o[i] = s;
    }
    f32x4 r0 = { o[0], o[1], o[2], o[3] };
    f32x4 r1 = { o[4], o[5], o[6], o[7] };
    __builtin_nontemporal_store(r0, (f32x4*)dst + tid);        // streamed once:
    __builtin_nontemporal_store(r1, (f32x4*)dst + tid + TPB);  // bypass caches
}

extern "C" void kernel_launch(void* const* d_in, const int* in_sizes, int n_in,
                              void* d_out, int out_size, void* d_ws, size_t ws_size,
                              hipStream_t stream) {
    const float* X = (const float*)d_in[0];   // (1,16,2048,2048) f32, contiguous
    float* Y = (float*)d_out;                 // same shape, f32
    (void)n_in; (void)d_ws; (void)ws_size; (void)out_size;

    const int n    = in_sizes[0];             // 16*2048*2048
    const int rows = n / ROW;                 // 32768 rows (dim = -1)
    spike_softmax_77360950935785_kernel<<<rows, TPB, 0, stream>>>(X, Y);
}